// Ultra_82678120448632
// MI455X (gfx1250) — compile-verified
//
#include <hip/hip_runtime.h>

typedef float v2f __attribute__((ext_vector_type(2)));
typedef float v4f __attribute__((ext_vector_type(4)));
typedef float v8f __attribute__((ext_vector_type(8)));

#define DD 64
#define TWO_D 128
#define LDA 132   // padded LDS stride for A tiles (conflict-free A-frag reads)
#define LDO 68    // padded LDS stride for output tile in lin_ln

// ---------------------------------------------------------------- utilities
__global__ void zero_f4_kernel(float4* __restrict__ p, int n4) {
    int i = blockIdx.x * blockDim.x + threadIdx.x;
    if (i < n4) p[i] = make_float4(0.f, 0.f, 0.f, 0.f);
}

__global__ void copy_f4_kernel(float4* __restrict__ dst, const float4* __restrict__ src, int n4) {
    int i = blockIdx.x * blockDim.x + threadIdx.x;
    if (i < n4) dst[i] = src[i];
}

// query[b] = rel_reps[b, r_index[b]]; boundary[b, h_index[b]] = query[b]
__global__ void set_query_kernel(const float* __restrict__ rel_reps,
                                 const int* __restrict__ h_index,
                                 const int* __restrict__ r_index,
                                 float* __restrict__ query,
                                 float* __restrict__ boundary,
                                 int N, int R) {
    int b = blockIdx.x;
    int d = threadIdx.x;                      // 64 threads
    float q = rel_reps[(b * R + r_index[b]) * DD + d];
    query[b * DD + d] = q;
    boundary[((long)b * N + h_index[b]) * DD + d] = q;
}

// ------------------------------------------------- per-layer relation MLP
// rel = relu(rel_reps @ pw1 + pb1) @ pw2 + pb2     (B*R rows, trivial cost)
__global__ __launch_bounds__(64) void rel_proj_kernel(
        const float* __restrict__ rel_reps,
        const float* __restrict__ pw1, const float* __restrict__ pb1,
        const float* __restrict__ pw2, const float* __restrict__ pb2,
        float* __restrict__ rel) {
    __shared__ float s_in[DD];
    __shared__ float s_hid[DD];
    int row = blockIdx.x;
    int t = threadIdx.x;
    s_in[t] = rel_reps[row * DD + t];
    __syncthreads();
    float acc = pb1[t];
    #pragma unroll 8
    for (int d = 0; d < DD; ++d) acc = fmaf(s_in[d], pw1[d * DD + t], acc);
    s_hid[t] = fmaxf(acc, 0.f);
    __syncthreads();
    float acc2 = pb2[t];
    #pragma unroll 8
    for (int h = 0; h < DD; ++h) acc2 = fmaf(s_hid[h], pw2[h * DD + t], acc2);
    rel[row * DD + t] = acc2;
}

// ------------------------------------------- distmult message + scatter-add
// agg[b, dst[e], d] += x[b, src[e], d] * rel[b, etype[e], d]
__global__ __launch_bounds__(256) void scatter_msg_kernel(
        const float* __restrict__ x, const float* __restrict__ rel,
        const int* __restrict__ src, const int* __restrict__ dst,
        const int* __restrict__ etype, float* __restrict__ agg,
        int total, int N, int R, int B) {
    int idx = blockIdx.x * 256 + threadIdx.x;
    if (idx >= total) return;
    int e = idx >> 6;
    int d = idx & 63;
    int s  = src[e];
    int t  = dst[e];
    int ty = etype[e];
    int xs = s * DD + d;
    int as = t * DD + d;
    int rs = ty * DD + d;
    for (int b = 0; b < B; ++b) {
        float m = x[b * N * DD + xs] * rel[b * R * DD + rs];
        unsafeAtomicAdd(&agg[b * N * DD + as], m);   // global_atomic_add_f32
    }
}

// ---------------- fused linear([x,agg]) + bias + LayerNorm + ReLU + residual
// 256 threads = 8 waves; block tile = 32 rows x 64 cols; K = 128
__global__ __launch_bounds__(256) void lin_ln_kernel(
        const float* __restrict__ x, const float* __restrict__ agg,
        const float* __restrict__ lw,  const float* __restrict__ lb,
        const float* __restrict__ lng, const float* __restrict__ lnb,
        float* __restrict__ xout) {
    __shared__ float sA[32 * LDA];
    __shared__ float sO[32 * LDO];
    int tid  = threadIdx.x;
    int wave = tid >> 5;
    int lane = tid & 31;
    int mbase = blockIdx.x * 32;

    // Stage A tile: 32 rows x 128 cols = [x row | agg row]
    {
        int row  = tid >> 3;
        int quad = tid & 7;
        int col0 = quad * 16;
        const float* srcp = (quad < 4) ? (x   + (mbase + row) * DD + col0)
                                       : (agg + (mbase + row) * DD + (col0 - DD));
        float* dstp = &sA[row * LDA + col0];
        *(v4f*)(dstp + 0)  = *(const v4f*)(srcp + 0);
        *(v4f*)(dstp + 4)  = *(const v4f*)(srcp + 4);
        *(v4f*)(dstp + 8)  = *(const v4f*)(srcp + 8);
        *(v4f*)(dstp + 12) = *(const v4f*)(srcp + 12);
    }
    __syncthreads();

    int mtile = wave >> 2;              // 0..1
    int ntile = wave & 3;               // 0..3
    int n     = lane & 15;
    int col   = ntile * 16 + n;
    int khalf = (lane >> 4) << 1;       // 0 or 2
    int mloc  = mtile * 16 + n;         // A-frag row index

    v8f c = {};
    #pragma unroll
    for (int k0 = 0; k0 < TWO_D; k0 += 4) {
        int kk = k0 + khalf;
        v2f a = *(const v2f*)&sA[mloc * LDA + kk];
        v2f b;
        b.x = lw[kk * DD + col];
        b.y = lw[(kk + 1) * DD + col];
        c = __builtin_amdgcn_wmma_f32_16x16x4_f32(false, a, false, b,
                                                  (short)0, c, false, false);
    }

    float bias = lb[col];
    int rbase = mtile * 16 + ((lane >> 4) << 3);
    #pragma unroll
    for (int j = 0; j < 8; ++j)
        sO[(rbase + j) * LDO + col] = c[j] + bias;
    __syncthreads();

    // LayerNorm over 64 cols per row (8 threads/row), ReLU, + residual x
    int r = tid >> 3;                   // 0..31
    int g = tid & 7;                    // col group of 8
    float v[8];
    float s = 0.f, s2 = 0.f;
    #pragma unroll
    for (int i = 0; i < 8; ++i) {
        v[i] = sO[r * LDO + g * 8 + i];
        s += v[i];
        s2 += v[i] * v[i];
    }
    s  += __shfl_xor(s, 1);  s  += __shfl_xor(s, 2);  s  += __shfl_xor(s, 4);
    s2 += __shfl_xor(s2, 1); s2 += __shfl_xor(s2, 2); s2 += __shfl_xor(s2, 4);
    float mu  = s * (1.f / 64.f);
    float var = s2 * (1.f / 64.f) - mu * mu;
    float rsi = rsqrtf(var + 1e-5f);
    #pragma unroll
    for (int i = 0; i < 8; ++i) {
        int cc = g * 8 + i;
        float o = (v[i] - mu) * rsi * lng[cc] + lnb[cc];
        o = fmaxf(o, 0.f) + sA[r * LDA + cc];   // residual: x row cached in sA[0..63]
        xout[(mbase + r) * DD + cc] = o;
    }
}

// ----------------------- final MLP: relu([x, query] @ mlp_w + mlp_b) -> out
// 256 threads = 8 waves; block tile = 16 rows x 128 cols; K = 128
__global__ __launch_bounds__(256) void final_mlp_kernel(
        const float* __restrict__ x, const float* __restrict__ query,
        const float* __restrict__ mw, const float* __restrict__ mb,
        float* __restrict__ out, int N) {
    __shared__ float sA[16 * LDA];
    int tid  = threadIdx.x;
    int wave = tid >> 5;
    int lane = tid & 31;
    int mbase = blockIdx.x * 16;
    int b = mbase / N;                  // 16 | N, block never straddles batches

    {
        int row  = tid >> 4;            // 0..15
        int seg  = tid & 15;            // col group of 8
        int col0 = seg * 8;
        const float* srcp = (seg < 8) ? (x + (mbase + row) * DD + col0)
                                      : (query + b * DD + (col0 - DD));
        float* dstp = &sA[row * LDA + col0];
        *(v4f*)(dstp + 0) = *(const v4f*)(srcp + 0);
        *(v4f*)(dstp + 4) = *(const v4f*)(srcp + 4);
    }
    __syncthreads();

    int ntile = wave;                   // 0..7
    int n     = lane & 15;
    int col   = ntile * 16 + n;
    int khalf = (lane >> 4) << 1;

    v8f c = {};
    #pragma unroll
    for (int k0 = 0; k0 < TWO_D; k0 += 4) {
        int kk = k0 + khalf;
        v2f a = *(const v2f*)&sA[n * LDA + kk];
        v2f bb;
        bb.x = mw[kk * TWO_D + col];
        bb.y = mw[(kk + 1) * TWO_D + col];
        c = __builtin_amdgcn_wmma_f32_16x16x4_f32(false, a, false, bb,
                                                  (short)0, c, false, false);
    }

    float bias = mb[col];
    int rbase = (lane >> 4) << 3;
    #pragma unroll
    for (int j = 0; j < 8; ++j) {
        float o = fmaxf(c[j] + bias, 0.f);
        out[(mbase + rbase + j) * TWO_D + col] = o;
    }
}

// --------------------------------------------------------------------- host
extern "C" void kernel_launch(void* const* d_in, const int* in_sizes, int n_in,
                              void* d_out, int out_size, void* d_ws, size_t ws_size,
                              hipStream_t stream) {
    const float* rel_reps = (const float*)d_in[0];
    const int*   h_index  = (const int*)d_in[1];
    const int*   r_index  = (const int*)d_in[2];
    const int*   edge_idx = (const int*)d_in[3];
    const int*   edge_typ = (const int*)d_in[4];
    const float* proj_w1  = (const float*)d_in[5];
    const float* proj_b1  = (const float*)d_in[6];
    const float* proj_w2  = (const float*)d_in[7];
    const float* proj_b2  = (const float*)d_in[8];
    const float* lin_w    = (const float*)d_in[9];
    const float* lin_b    = (const float*)d_in[10];
    const float* ln_g     = (const float*)d_in[11];
    const float* ln_b     = (const float*)d_in[12];
    const float* mlp_w    = (const float*)d_in[13];
    const float* mlp_b    = (const float*)d_in[14];

    const int B = in_sizes[1];
    const int E = in_sizes[4];
    const int R = in_sizes[0] / (B * DD);
    const int N = (int)((long)out_size / ((long)B * TWO_D));
    const int BN = B * N;
    const int L = 6;

    // workspace layout (floats, all 256B aligned)
    float* ws = (float*)d_ws;
    size_t off = 0;
    float* query    = ws + off; off += (size_t)B * DD;
    float* boundary = ws + off; off += (size_t)BN * DD;
    float* xA       = ws + off; off += (size_t)BN * DD;
    float* xB       = ws + off; off += (size_t)BN * DD;
    float* agg      = ws + off; off += (size_t)BN * DD;
    float* rel      = ws + off; off += (size_t)B * R * DD;
    (void)ws_size; (void)n_in;

    const int bigN4 = BN * DD / 4;
    const int cpyBlocks = (bigN4 + 255) / 256;

    // boundary = 0; boundary[b, h] = query; x0 = boundary
    zero_f4_kernel<<<cpyBlocks, 256, 0, stream>>>((float4*)boundary, bigN4);
    set_query_kernel<<<B, DD, 0, stream>>>(rel_reps, h_index, r_index, query,
                                           boundary, N, R);
    copy_f4_kernel<<<cpyBlocks, 256, 0, stream>>>((float4*)xA, (const float4*)boundary, bigN4);

    float* x  = xA;
    float* xn = xB;
    const int scatterTotal  = E * DD;
    const int scatterBlocks = (scatterTotal + 255) / 256;

    for (int l = 0; l < L; ++l) {
        rel_proj_kernel<<<B * R, DD, 0, stream>>>(
            rel_reps,
            proj_w1 + (size_t)l * DD * DD, proj_b1 + (size_t)l * DD,
            proj_w2 + (size_t)l * DD * DD, proj_b2 + (size_t)l * DD,
            rel);
        copy_f4_kernel<<<cpyBlocks, 256, 0, stream>>>((float4*)agg, (const float4*)boundary, bigN4);
        scatter_msg_kernel<<<scatterBlocks, 256, 0, stream>>>(
            x, rel, edge_idx, edge_idx + E, edge_typ, agg, scatterTotal, N, R, B);
        lin_ln_kernel<<<BN / 32, 256, 0, stream>>>(
            x, agg,
            lin_w + (size_t)l * TWO_D * DD, lin_b + (size_t)l * DD,
            ln_g + (size_t)l * DD, ln_b + (size_t)l * DD,
            xn);
        float* tmp = x; x = xn; xn = tmp;
    }

    final_mlp_kernel<<<BN / 16, 256, 0, stream>>>(
        x, query, mlp_w, mlp_b, (float*)d_out, N);
}